// SNNContinuousController_45062796869840
// MI455X (gfx1250) — compile-verified
//
#include <hip/hip_runtime.h>
#include <math.h>

// ---------------------------------------------------------------------------
// Fused SNN (fc1 -> LIF scan -> spike record -> mean -> fc2 -> tanh)
// MI455X / gfx1250, wave32.
//
// Roofline: read x (73.7MB) + write spk_rec (786MB) ~= 860MB -> ~37us @23.3TB/s.
// Single pass: cur[t] is computed on the fly with V_WMMA_F32_16X16X4_F32
// (K=9 padded to 12 -> 3 chained WMMAs), membrane state lives in VGPRs for
// all 1000 steps, spikes stored non-temporally (write-once, keep L2 for x).
// ---------------------------------------------------------------------------

typedef float v2f __attribute__((ext_vector_type(2)));
typedef float v8f __attribute__((ext_vector_type(8)));

constexpr int   T_STEPS = 1000;
constexpr int   BATCH   = 2048;
constexpr int   D_IN    = 9;
constexpr int   HID     = 96;
constexpr float BETA    = 0.92f;
constexpr float THRESH  = 1.0f;

#if defined(__has_builtin)
#if __has_builtin(__builtin_amdgcn_wmma_f32_16x16x4_f32)
#define HAVE_WMMA_F32X4 1
#endif
#endif
#ifndef HAVE_WMMA_F32X4
#define HAVE_WMMA_F32X4 0
#endif

// A-fragment loader for one lane: 16x4 f32 A-matrix layout is
//   lanes 0-15:  M=lane,    VGPR0=K0, VGPR1=K1
//   lanes 16-31: M=lane-16, VGPR0=K2, VGPR1=K3
// p points at x[t, b0+m, 0]; chunk c covers k = 4c .. 4c+3; k>=9 is zero-pad.
__device__ __forceinline__ void load_a(const float* __restrict__ p, int half,
                                       v2f a[3]) {
  const int o = 2 * half;
  a[0].x = p[0 + o];                       // k = 0/1  (half 0) or 2/3  (half 1)
  a[0].y = p[1 + o];
  a[1].x = p[4 + o];                       // k = 4/5  or 6/7
  a[1].y = p[5 + o];
  a[2].x = (half == 0) ? p[8] : 0.0f;      // k = 8 valid only; 9..11 padded
  a[2].y = 0.0f;
}

__global__ __launch_bounds__(192) void snn_lif_kernel(
    const float* __restrict__ X,      // [T, B, 9]
    const float* __restrict__ W1,     // [96, 9]
    const float* __restrict__ b1,     // [96]
    const float* __restrict__ W2,     // [1, 96]
    float* __restrict__ spk_out,      // [T, B, 96]
    float* __restrict__ steer_acc)    // [B], pre-zeroed accumulator
{
  const int wave = threadIdx.x >> 5;     // 0..5  -> h-tile within block
  const int lane = threadIdx.x & 31;
  const int half = lane >> 4;            // 0/1 (lane group)
  const int n    = lane & 15;            // tile column index
  const int b0   = blockIdx.x * 16;      // b-tile (128 blocks)
  const int h0   = wave * 16;
  const int h    = h0 + n;               // this lane's h column (fixed)

  // ---- B operand: W1^T as 4x16 f32 fragments, 3 K-chunks, loop-invariant.
  //   lanes 0-15:  N=lane,    VGPR0=K0, VGPR1=K1
  //   lanes 16-31: N=lane-16, VGPR0=K2, VGPR1=K3
  v2f bmat[3];
#pragma unroll
  for (int c = 0; c < 3; ++c) {
    const int k = 4 * c + 2 * half;
    bmat[c].x = (k     < D_IN) ? W1[h * D_IN + k]     : 0.0f;
    bmat[c].y = (k + 1 < D_IN) ? W1[h * D_IN + k + 1] : 0.0f;
  }

  const float bias = b1[h];
  const float w2h  = W2[h];

  // C/D layout: VGPR r, this lane -> (b = b0 + r + 8*half, h = h0 + n).
  float mem[8], spkSum[8];
#pragma unroll
  for (int r = 0; r < 8; ++r) { mem[r] = 0.0f; spkSum[r] = 0.0f; }

  const float* xrow = X + (size_t)(b0 + n) * D_IN;   // A-row for this lane
  const size_t xstep = (size_t)BATCH * D_IN;

  v2f a_cur[3], a_nxt[3];
  load_a(xrow, half, a_cur);
#pragma unroll
  for (int c = 0; c < 3; ++c) a_nxt[c] = a_cur[c];

  for (int t = 0; t < T_STEPS; ++t) {
    // Software pipeline: issue next timestep's A loads before this step's
    // LIF math + stores so load latency overlaps useful work.
    if (t + 1 < T_STEPS) load_a(xrow + (size_t)(t + 1) * xstep, half, a_nxt);

    v8f acc = {0.f, 0.f, 0.f, 0.f, 0.f, 0.f, 0.f, 0.f};
#if HAVE_WMMA_F32X4
    acc = __builtin_amdgcn_wmma_f32_16x16x4_f32(false, a_cur[0], false, bmat[0],
                                                (short)0, acc, false, false);
    acc = __builtin_amdgcn_wmma_f32_16x16x4_f32(false, a_cur[1], false, bmat[1],
                                                (short)0, acc, false, false);
    acc = __builtin_amdgcn_wmma_f32_16x16x4_f32(false, a_cur[2], false, bmat[2],
                                                (short)0, acc, false, false);
#else
    // Exact scalar fallback (signals wmma=0 in the histogram -> fix builtin).
#pragma unroll
    for (int r = 0; r < 8; ++r) {
      const float* xp = X + (size_t)t * xstep + (size_t)(b0 + r + 8 * half) * D_IN;
      float s = 0.0f;
      for (int k = 0; k < D_IN; ++k) s = __builtin_fmaf(xp[k], W1[h * D_IN + k], s);
      acc[r] = s;
    }
#endif

    // LIF: reset from previous mem, integrate, fire. Matches snntorch Leaky
    // subtract-reset: mem' = beta*mem + cur - (mem>thr)*thr; spk = mem' > thr.
    const size_t tbase = (size_t)t * BATCH * HID;
#pragma unroll
    for (int r = 0; r < 8; ++r) {
      const float cur   = acc[r] + bias;
      const float reset = (mem[r] > THRESH) ? THRESH : 0.0f;
      const float mnew  = __builtin_fmaf(BETA, mem[r], cur) - reset;
      mem[r] = mnew;
      const float spk = (mnew > THRESH) ? 1.0f : 0.0f;
      spkSum[r] += spk;
      __builtin_nontemporal_store(
          spk, spk_out + tbase + (size_t)(b0 + r + 8 * half) * HID + h);
    }

#pragma unroll
    for (int c = 0; c < 3; ++c) a_cur[c] = a_nxt[c];
  }

  // ---- steer partials: features = spkSum/T, fold in W2, reduce over the 16
  // h-columns held by this half-wave (xor masks 1..8 never cross bit 4).
  const float scale = w2h * (1.0f / (float)T_STEPS);
#pragma unroll
  for (int r = 0; r < 8; ++r) {
    float v = spkSum[r] * scale;
    v += __shfl_xor(v, 8, 32);
    v += __shfl_xor(v, 4, 32);
    v += __shfl_xor(v, 2, 32);
    v += __shfl_xor(v, 1, 32);
    if (n == 0) atomicAdd(&steer_acc[b0 + r + 8 * half], v);
  }
}

__global__ void steer_finish_kernel(const float* __restrict__ acc,
                                    const float* __restrict__ b2,
                                    float* __restrict__ steer) {
  const int b = blockIdx.x * blockDim.x + threadIdx.x;
  if (b < BATCH) steer[b] = tanhf(acc[b] + b2[0]);
}

extern "C" void kernel_launch(void* const* d_in, const int* in_sizes, int n_in,
                              void* d_out, int out_size, void* d_ws, size_t ws_size,
                              hipStream_t stream) {
  (void)in_sizes; (void)n_in; (void)out_size; (void)ws_size;
  const float* X  = (const float*)d_in[0];   // [1000,2048,9]
  const float* W1 = (const float*)d_in[1];   // [96,9]
  const float* b1 = (const float*)d_in[2];   // [96]
  const float* W2 = (const float*)d_in[3];   // [1,96]
  const float* b2 = (const float*)d_in[4];   // [1]

  float* out   = (float*)d_out;
  float* steer = out;            // [2048]        (tuple element 0)
  float* spk   = out + BATCH;    // [1000,2048,96] (tuple element 1)
  float* acc   = (float*)d_ws;   // [2048] partial-sum accumulator

  hipMemsetAsync(acc, 0, BATCH * sizeof(float), stream);

  // 128 blocks x 6 waves: block = one b-tile (shared x slice), wave = h-tile.
  snn_lif_kernel<<<dim3(BATCH / 16), dim3(192), 0, stream>>>(X, W1, b1, W2,
                                                             spk, acc);
  steer_finish_kernel<<<dim3((BATCH + 255) / 256), dim3(256), 0, stream>>>(
      acc, b2, steer);
}